// DTranNER_20650202759782
// MI455X (gfx1250) — compile-verified
//
#include <hip/hip_runtime.h>

#define Bn 256
#define Tn 512
#define Kn 24
#define KSTART 22
#define KSTOP 23
#define FNEG -100.0f

#define RING 16   // LDS tile ring (power of two)
#define DEPTH 15  // prefetch distance; wait leaves DEPTH newest in flight

typedef __attribute__((ext_vector_type(2))) float v2f;
typedef __attribute__((ext_vector_type(8))) float v8f;
typedef __attribute__((ext_vector_type(4))) unsigned int v4u;
typedef __attribute__((ext_vector_type(4))) int v4i;
typedef __attribute__((ext_vector_type(8))) int v8i;

#if __has_builtin(__builtin_amdgcn_tensor_load_to_lds) && __has_builtin(__builtin_amdgcn_s_wait_tensorcnt)
#define USE_TDM 1
#if __has_include(<hip/amd_detail/amd_gfx1250_TDM.h>)
#define TDM_6ARG 1
#endif
#endif

// Hard check on the DEVICE pass only (__HIP_DEVICE_COMPILE__): if the TDM
// builtins are missing from the gfx1250 compilation, fail loudly.  The host
// pass (which lacks amdgcn builtins but may still define __AMDGCN__ on this
// toolchain) keeps the plain-HIP fallback below so hipcc can parse.
#if defined(__HIP_DEVICE_COMPILE__) && !defined(USE_TDM)
#error "gfx1250 device pass: tensor_load_to_lds / s_wait_tensorcnt unavailable"
#endif

#ifdef USE_TDM
// 1-D tile copy: 576 f32 (2304 B) global -> LDS, tracked on TENSORcnt.
// D# group0: count=1 | lds_addr | global_addr[56:0] | type=2
// D# group1: data_size=4B, tensor_dim0=tile_dim0=576, dim1=1, stride0=576
__device__ __forceinline__ void tdm_tile_load(unsigned lds_byte, const void* gaddr) {
  unsigned long long ga = (unsigned long long)(size_t)gaddr;
  v4u g0 = { 1u, lds_byte, (unsigned)ga,
             ((unsigned)((ga >> 32) & 0x01FFFFFFull)) | 0x80000000u };
  v8i g1 = { (int)(2u << 16),      // data_size = 2 (4 bytes)
             (int)(576u << 16),    // tensor_dim0 low16
             (int)(1u << 16),      // tensor_dim0 hi=0 | tensor_dim1 = 1
             (int)(576u << 16),    // tile_dim0 = 576
             0,                    // tile_dim1/2 unused
             576,                  // tensor_dim0_stride
             0, 0 };
  v4i z4 = { 0, 0, 0, 0 };
#ifdef TDM_6ARG
  v8i z8 = { 0, 0, 0, 0, 0, 0, 0, 0 };
  __builtin_amdgcn_tensor_load_to_lds(g0, g1, z4, z4, z8, 0);
#else
  __builtin_amdgcn_tensor_load_to_lds(g0, g1, z4, z4, 0);
#endif
}
#endif

// ---------------------------------------------------------------------------
// Kernel 1: standard CRF forward (alpha) via WMMA f32 16x16x4, plus gold
// linear score.  One wave handles 16 batch chains: S(24x16) = expT(24x24) @ W.
// Writes out[b] = alpha[b] - score[b].
// ---------------------------------------------------------------------------
__global__ __launch_bounds__(32) void crf_linear_kernel(
    const float* __restrict__ feats, const float* __restrict__ trans,
    const int* __restrict__ tags, float* __restrict__ out) {
  __shared__ __attribute__((aligned(16))) float Wt[16 * 26];  // [b][j], padded
  const int lane = (int)threadIdx.x;
  const int n = lane & 15;   // batch column within tile / A-matrix row
  const int hi = lane >> 4;  // lane half selects K / M sub-assignment
  const int bbase = (int)blockIdx.x * 16;

  // Constant A fragments: exp(transitions[j,i]); rows j>=24 padded with 0.
  // A 16x4 layout: lanes hold M=lane&15; VGPR0 = K0/K2, VGPR1 = K1/K3.
  v2f afrag[2][6];
#pragma unroll
  for (int mt = 0; mt < 2; ++mt) {
    const int row = mt * 16 + n;
#pragma unroll
    for (int kc = 0; kc < 6; ++kc) {
      const int c0 = kc * 4 + hi * 2;
      float x0 = 0.0f, x1 = 0.0f;
      if (row < Kn) {
        x0 = __expf(trans[row * Kn + c0]);
        x1 = __expf(trans[row * Kn + c0 + 1]);
      }
      v2f a; a.x = x0; a.y = x1;
      afrag[mt][kc] = a;
    }
  }

  // exp(transitions[STOP, j]) in D layout for the final reduction.
  float ets0[8], ets1[8];
#pragma unroll
  for (int v = 0; v < 8; ++v) {
    const int j0 = v + hi * 8;
    const int j1 = 16 + v + hi * 8;
    ets0[v] = __expf(trans[KSTOP * Kn + j0]);
    ets1[v] = (j1 < Kn) ? __expf(trans[KSTOP * Kn + j1]) : 0.0f;
  }

  // fv in WMMA D layout: VGPR v holds state j=v / j=v+8 (lane halves).
  float fv0[8], fv1[8];
#pragma unroll
  for (int v = 0; v < 8; ++v) { fv0[v] = FNEG; fv1[v] = FNEG; }
  if (hi == 0) fv1[6] = 0.0f;  // START = 22 = tile1, M=6, lanes 0-15

  float mm;
  for (int t = 0; t < Tn; ++t) {
    // Emission loads issued early so they overlap the WMMA chain.
    const float* fb = feats + ((size_t)(bbase + n) * Tn + t) * Kn;
    float ft0[8], ft1[8];
    if (hi == 0) {
      float4 q0 = *(const float4*)(fb + 0);
      float4 q1 = *(const float4*)(fb + 4);
      float4 q2 = *(const float4*)(fb + 16);
      float4 q3 = *(const float4*)(fb + 20);
      ft0[0] = q0.x; ft0[1] = q0.y; ft0[2] = q0.z; ft0[3] = q0.w;
      ft0[4] = q1.x; ft0[5] = q1.y; ft0[6] = q1.z; ft0[7] = q1.w;
      ft1[0] = q2.x; ft1[1] = q2.y; ft1[2] = q2.z; ft1[3] = q2.w;
      ft1[4] = q3.x; ft1[5] = q3.y; ft1[6] = q3.z; ft1[7] = q3.w;
    } else {
      float4 q0 = *(const float4*)(fb + 8);
      float4 q1 = *(const float4*)(fb + 12);
      ft0[0] = q0.x; ft0[1] = q0.y; ft0[2] = q0.z; ft0[3] = q0.w;
      ft0[4] = q1.x; ft0[5] = q1.y; ft0[6] = q1.z; ft0[7] = q1.w;
#pragma unroll
      for (int v = 0; v < 8; ++v) ft1[v] = 0.0f;
    }

    // Per-chain max over states (lane n pairs with lane n+16).
    mm = FNEG;
#pragma unroll
    for (int v = 0; v < 8; ++v) { mm = fmaxf(mm, fv0[v]); mm = fmaxf(mm, fv1[v]); }
    mm = fmaxf(mm, __shfl_xor(mm, 16, 32));

    // W[b][j] = exp(fv - m), stored transposed in LDS so B-fragments load as
    // contiguous 64-bit pairs. Same-wave LDS ops are hardware in-order.
#pragma unroll
    for (int v = 0; v < 8; ++v) Wt[n * 26 + (v + hi * 8)] = __expf(fv0[v] - mm);
    if (hi == 0) {
#pragma unroll
      for (int v = 0; v < 8; ++v) Wt[n * 26 + 16 + v] = __expf(fv1[v] - mm);
    }

    v8f acc0 = {};
    v8f acc1 = {};
#pragma unroll
    for (int kc = 0; kc < 6; ++kc) {
      v2f bfrag = *(const v2f*)&Wt[n * 26 + kc * 4 + hi * 2];
      acc0 = __builtin_amdgcn_wmma_f32_16x16x4_f32(
          false, afrag[0][kc], false, bfrag, (short)0, acc0, false, false);
      acc1 = __builtin_amdgcn_wmma_f32_16x16x4_f32(
          false, afrag[1][kc], false, bfrag, (short)0, acc1, false, false);
    }

    // Branch-free update (lanes with hi==1 compute a harmless value for fv1
    // and select FNEG; avoids exec-masked log blocks in the inner loop).
#pragma unroll
    for (int v = 0; v < 8; ++v) {
      fv0[v] = ft0[v] + mm + __logf(fmaxf(acc0[v], 1e-37f));
      const float nv1 = ft1[v] + mm + __logf(fmaxf(acc1[v], 1e-37f));
      fv1[v] = (hi == 0) ? nv1 : FNEG;
    }
  }

  // alpha = m + log( sum_j exp(trans[STOP,j]) * exp(fv[j]-m) )
  mm = FNEG;
#pragma unroll
  for (int v = 0; v < 8; ++v) { mm = fmaxf(mm, fv0[v]); mm = fmaxf(mm, fv1[v]); }
  mm = fmaxf(mm, __shfl_xor(mm, 16, 32));
  float part = 0.0f;
#pragma unroll
  for (int v = 0; v < 8; ++v)
    part += ets0[v] * __expf(fv0[v] - mm) + ets1[v] * __expf(fv1[v] - mm);
  part += __shfl_xor(part, 16, 32);
  const float alpha = mm + __logf(fmaxf(part, 1e-37f));

  // Gold linear score, one chain per lane (lanes 0-15).
  if (hi == 0) {
    const int b = bbase + n;
    float sc = 0.0f;
    int prev = KSTART;
    for (int t = 0; t < Tn; ++t) {
      const int nx = tags[b * Tn + t];
      sc += trans[nx * Kn + prev] + feats[((size_t)b * Tn + t) * Kn + nx];
      prev = nx;
    }
    sc += trans[KSTOP * Kn + prev];
    out[b] = alpha - sc;
  }
}

// ---------------------------------------------------------------------------
// Kernel 2: pairwise CRF forward.  One single-wave workgroup per chain b.
// A 16-entry LDS tile ring is fed by TDM (tensor_load_to_lds) with prefetch
// distance 15; TENSORcnt completes in order per wave, so
// s_wait_tensorcnt(DEPTH) guarantees tile t is resident while 15 future tiles
// stream.  Gold pairwise score gathers from the resident tile.
// out[b] += alpha_pp - score_pp.
// ---------------------------------------------------------------------------
__global__ __launch_bounds__(32) void crf_pairwise_kernel(
    const float* __restrict__ fpp, const int* __restrict__ tags,
    float* __restrict__ out) {
  __shared__ __attribute__((aligned(16))) float smem[RING * 576 + 32];
  const int lane = (int)threadIdx.x;
  const int b = (int)blockIdx.x;
  const int rj = (lane < Kn) ? lane : (Kn - 1);
  const unsigned wboff = (unsigned)(RING * 576);
  const float* gb = fpp + (size_t)b * Tn * (Kn * Kn);

  // Prologue: fill DEPTH ring slots.
#ifdef USE_TDM
#pragma unroll
  for (int p = 0; p < DEPTH; ++p)
    tdm_tile_load((unsigned)p * 576u * 4u, gb + (size_t)p * 576);
#else
  for (int p = 0; p < DEPTH; ++p) {
    const float* s = gb + (size_t)p * 576;
    float* d = &smem[p * 576];
    *(float4*)(d + lane * 4) = *(const float4*)(s + lane * 4);
    *(float4*)(d + (lane + 32) * 4) = *(const float4*)(s + (lane + 32) * 4);
    *(float4*)(d + (lane + 64) * 4) = *(const float4*)(s + (lane + 64) * 4);
    *(float4*)(d + (lane + 96) * 4) = *(const float4*)(s + (lane + 96) * 4);
    if (lane < 16)
      *(float4*)(d + (lane + 128) * 4) = *(const float4*)(s + (lane + 128) * 4);
  }
#endif

  float fv2 = (lane == KSTART) ? 0.0f : FNEG;
  float spp = 0.0f;
  int prev = KSTART;

  for (int t = 0; t < Tn - 1; ++t) {
    const unsigned cb = (unsigned)(t & (RING - 1)) * 576u;
#ifdef USE_TDM
    if (t + DEPTH < Tn) {
      // Writes slot (t-1)&15, consumed last iteration -> no WAR hazard.
      tdm_tile_load((unsigned)((t + DEPTH) & (RING - 1)) * 576u * 4u,
                    gb + (size_t)(t + DEPTH) * 576);
      __builtin_amdgcn_s_wait_tensorcnt(DEPTH);  // tile t guaranteed resident
    } else {
      __builtin_amdgcn_s_wait_tensorcnt(0);      // tail: everything issued
    }
#else
    if (t + DEPTH < Tn) {
      const float* s = gb + (size_t)(t + DEPTH) * 576;
      float* d = &smem[((t + DEPTH) & (RING - 1)) * 576];
      *(float4*)(d + lane * 4) = *(const float4*)(s + lane * 4);
      *(float4*)(d + (lane + 32) * 4) = *(const float4*)(s + (lane + 32) * 4);
      *(float4*)(d + (lane + 64) * 4) = *(const float4*)(s + (lane + 64) * 4);
      *(float4*)(d + (lane + 96) * 4) = *(const float4*)(s + (lane + 96) * 4);
      if (lane < 16)
        *(float4*)(d + (lane + 128) * 4) = *(const float4*)(s + (lane + 128) * 4);
    }
#endif

    // nv[j] = m + log( sum_i exp(fpp[j,i]) * exp(fv2[i]-m) )
    float m2 = fv2;
#pragma unroll
    for (int off = 16; off > 0; off >>= 1)
      m2 = fmaxf(m2, __shfl_xor(m2, off, 32));
    const float w2 = __expf(fv2 - m2);
    smem[wboff + (unsigned)lane] = w2;  // same-wave LDS: in-order, no barrier
    const float* row = &smem[cb + (unsigned)rj * Kn];
    const float* wv = &smem[wboff];
    float a0 = 0.f, a1 = 0.f, a2 = 0.f, a3 = 0.f;
#pragma unroll
    for (int c = 0; c < 6; ++c) {
      float4 r4 = *(const float4*)(row + c * 4);
      float4 w4 = *(const float4*)(wv + c * 4);
      a0 = fmaf(__expf(r4.x), w4.x, a0);
      a1 = fmaf(__expf(r4.y), w4.y, a1);
      a2 = fmaf(__expf(r4.z), w4.z, a2);
      a3 = fmaf(__expf(r4.w), w4.w, a3);
    }
    const float s2 = (a0 + a1) + (a2 + a3);
    if (lane < Kn) fv2 = m2 + __logf(fmaxf(s2, 1e-37f));

    // Gold pairwise score gather from the resident tile.
    if (lane == 0) {
      const int nx = tags[b * Tn + t];
      spp += smem[cb + (unsigned)nx * Kn + (unsigned)prev];
      prev = nx;
    }
  }

#ifdef USE_TDM
  __builtin_amdgcn_s_wait_tensorcnt(0);
#endif
  // Final tile (t = Tn-1); only row STOP is consumed.
  const float* last = &smem[(unsigned)((Tn - 1) & (RING - 1)) * 576u];
  float m2 = fv2;
#pragma unroll
  for (int off = 16; off > 0; off >>= 1)
    m2 = fmaxf(m2, __shfl_xor(m2, off, 32));
  const float w2 = __expf(fv2 - m2);
  float term = (lane < Kn) ? __expf(last[KSTOP * Kn + lane]) * w2 : 0.0f;
#pragma unroll
  for (int off = 16; off > 0; off >>= 1)
    term += __shfl_xor(term, off, 32);
  const float alpha_pp = m2 + __logf(fmaxf(term, 1e-37f));

  if (lane == 0) {
    const int lastTag = tags[b * Tn + (Tn - 1)];
    spp += last[KSTOP * Kn + lastTag];
    out[b] = out[b] + alpha_pp - spp;  // out held alpha - score from kernel 1
  }
}

extern "C" void kernel_launch(void* const* d_in, const int* in_sizes, int n_in,
                              void* d_out, int out_size, void* d_ws, size_t ws_size,
                              hipStream_t stream) {
  const float* feats = (const float*)d_in[0];
  const float* fpp   = (const float*)d_in[1];
  const float* trans = (const float*)d_in[2];
  const int*   tags  = (const int*)d_in[3];
  float* out = (float*)d_out;
  (void)in_sizes; (void)n_in; (void)out_size; (void)d_ws; (void)ws_size;

  crf_linear_kernel<<<Bn / 16, 32, 0, stream>>>(feats, trans, tags, out);
  crf_pairwise_kernel<<<Bn, 32, 0, stream>>>(fpp, tags, out);
}